// VectorQuantizer_9783935500409
// MI455X (gfx1250) — compile-verified
//
#include <hip/hip_runtime.h>

// ---------------------------------------------------------------------------
// VQ forward for MI455X (gfx1250, wave32).
//   x: (F,B,D) f32, w: (F,D,K) f32 ; F=32,B=4096,D=64,K=512
//   out = [ quantized (F,B,D) f32 , loss (1) f32 ]
// Cross-term GEMM on V_WMMA_F32_16X16X4_F32 (full fp32 -> exact argmin math),
// codebook staged in LDS, argmin via in-half shfl_xor reduction.
// ---------------------------------------------------------------------------

typedef __attribute__((ext_vector_type(2))) float v2f;
typedef __attribute__((ext_vector_type(4))) float v4f;
typedef __attribute__((ext_vector_type(8))) float v8f;

#define VQ_F 32
#define VQ_B 4096
#define VQ_D 64
#define VQ_K 512
#define ROWS_PER_WG 128       // 8 waves x 16 rows
#define THREADS 256

// LDS budget: 32*512 float2 (w pairs) + 512 f32 (wnorm) + 128 i32 (kidx)
#define SMEM_WPAIRS (32 * 512)
#define SMEM_BYTES  (SMEM_WPAIRS * 8 + VQ_K * 4 + ROWS_PER_WG * 4)

__global__ __launch_bounds__(THREADS)
void vq_wmma_kernel(const float* __restrict__ x, const float* __restrict__ w,
                    float* __restrict__ qout, float* __restrict__ loss) {
    extern __shared__ char smem[];
    v2f*   s_w    = (v2f*)smem;                               // [(d>>1)*512 + k] = (w[d][k], w[d+1][k])
    float* s_wn   = (float*)(smem + SMEM_WPAIRS * 8);         // [512] column norms
    int*   s_kidx = (int*)(smem + SMEM_WPAIRS * 8 + VQ_K*4);  // [128] winning code per row

    const int t    = threadIdx.x;
    const int lane = t & 31;
    const int wv   = t >> 5;          // wave id 0..7
    const int f    = blockIdx.y;
    const int bwg  = blockIdx.x * ROWS_PER_WG;

    const float* wf = w + (size_t)f * VQ_D * VQ_K;
    const float* xf = x + ((size_t)f * VQ_B + bwg) * VQ_D;

    // ---- stage codebook into LDS in (d-pair, k) float2 layout ----
    for (int i = 0; i < 32; ++i) {
        int flat = (i * THREADS + t) * 4;       // element index into 64x512
        int d = flat >> 9;
        int k = flat & 511;
        v4f v = *(const v4f*)(wf + flat);
        float* dst = (float*)s_w;
        int base = ((d >> 1) * 512) * 2 + (d & 1);
        dst[base + (k + 0) * 2] = v.x;
        dst[base + (k + 1) * 2] = v.y;
        dst[base + (k + 2) * 2] = v.z;
        dst[base + (k + 3) * 2] = v.w;
    }
    __syncthreads();

    // ---- per-code squared norms ||w_k||^2 ----
    for (int k = t; k < VQ_K; k += THREADS) {
        float acc = 0.f;
        for (int p = 0; p < 32; ++p) {
            v2f pv = s_w[p * 512 + k];
            acc += pv.x * pv.x + pv.y * pv.y;
        }
        s_wn[k] = acc;
    }
    __syncthreads();

    // ---- preload this wave's 16x64 x tile as WMMA A fragments ----
    // A 16x4 f32 layout: lanes 0-15 (M=0..15) hold K=0,1 ; lanes 16-31 hold K=2,3
    const int hi   = lane >> 4;       // lane half
    const int mrow = lane & 15;
    v2f a[16];
    #pragma unroll
    for (int s = 0; s < 16; ++s) {
        int d0 = s * 4 + hi * 2;
        a[s] = *(const v2f*)(xf + (wv * 16 + mrow) * VQ_D + d0);
    }

    float minv[8];
    int   mini[8];
    #pragma unroll
    for (int j = 0; j < 8; ++j) { minv[j] = 3.402823466e38f; mini[j] = 0; }

    // ---- GEMM (16x16 tiles, K=64 in 16 WMMA steps) + running argmin ----
    for (int tile = 0; tile < 32; ++tile) {
        int n = tile * 16 + mrow;     // code column owned by this lane
        v8f c = {};
        #pragma unroll
        for (int s = 0; s < 16; ++s) {
            int dp = s * 2 + hi;                  // (s*4 + hi*2) >> 1
            v2f bv = s_w[dp * 512 + n];           // B rows K=d0, d0+1 for col n
            c = __builtin_amdgcn_wmma_f32_16x16x4_f32(
                    false, a[s], false, bv, (short)0, c, false, false);
        }
        float wn = s_wn[n];
        #pragma unroll
        for (int j = 0; j < 8; ++j) {
            float dist = wn - 2.0f * c[j];        // ||x||^2 constant per row: drop
            if (dist < minv[j]) { minv[j] = dist; mini[j] = n; }
        }
    }

    // ---- argmin across the 16 lanes of each half (C: VGPR j <-> rows j / j+8) ----
    #pragma unroll
    for (int off = 1; off <= 8; off <<= 1) {
        #pragma unroll
        for (int j = 0; j < 8; ++j) {
            float ov = __shfl_xor(minv[j], off, 32);
            int   oi = __shfl_xor(mini[j], off, 32);
            if (ov < minv[j] || (ov == minv[j] && oi < mini[j])) {
                minv[j] = ov; mini[j] = oi;
            }
        }
    }
    #pragma unroll
    for (int j = 0; j < 8; ++j) {
        if (mrow == j) s_kidx[wv * 16 + hi * 8 + j] = mini[j];
    }
    __syncthreads();

    // ---- gather winning columns, write quantized, accumulate loss ----
    const float scale = 1.25f / (float)((size_t)VQ_F * VQ_B * VQ_D);
    float* qf = qout + ((size_t)f * VQ_B + bwg) * VQ_D;
    const float* wflat = (const float*)s_w;
    float sq = 0.f;
    for (int i = 0; i < 32; ++i) {
        int flat = i * THREADS + t;   // 128 rows x 64 d, coalesced
        int row  = flat >> 6;
        int d    = flat & 63;
        int kmin = s_kidx[row];
        float qv = wflat[(((d >> 1) * 512 + kmin) << 1) + (d & 1)];
        float xv = xf[flat];
        qf[flat] = qv;
        float diff = qv - xv;
        sq += diff * diff;
    }
    #pragma unroll
    for (int off = 16; off >= 1; off >>= 1) sq += __shfl_xor(sq, off, 32);
    if (lane == 0) atomicAdd(loss, sq * scale);
}

extern "C" void kernel_launch(void* const* d_in, const int* in_sizes, int n_in,
                              void* d_out, int out_size, void* d_ws, size_t ws_size,
                              hipStream_t stream) {
    const float* x = (const float*)d_in[0];
    const float* w = (const float*)d_in[1];
    float* qout = (float*)d_out;
    float* loss = qout + (size_t)VQ_F * VQ_B * VQ_D;

    hipMemsetAsync(loss, 0, sizeof(float), stream);   // capturable, stream-ordered

    dim3 grid(VQ_B / ROWS_PER_WG, VQ_F);
    vq_wmma_kernel<<<grid, THREADS, SMEM_BYTES, stream>>>(x, w, qout, loss);
}